// SmoothMask_8014408975108
// MI455X (gfx1250) — compile-verified
//
#include <hip/hip_runtime.h>
#include <hip/hip_bf16.h>

// ---------------------------------------------------------------------------
// SmoothMask pipeline for MI455X (gfx1250).
//
// Structure insight: the reference scatters ALL batches' keypoints into ONE
// (364,280) heatmap, so the mask is batch-independent. We compute the mask
// once in workspace, then do a single memory-bound float4 streaming pass
// over the 32x3x512x512 input (~201MB -> ~8.6us floor at 23.3 TB/s).
//
// The 31x31 Gaussian is separable (k = u (x) u). Each 1-D conv on a 16x16
// output tile is a (16x48)x(48x16) matmul with a banded-Toeplitz u-matrix,
// executed as 12 chained V_WMMA_F32_16X16X4_F32 ops (full fp32 precision,
// needed for the 1e-7 threshold).
//
// This revision makes all zero-pad guards branchless (clamped address +
// v_cndmask instead of EXEC-predicated loads) so EXEC stays all-1s through
// the WMMA chain, and uses non-temporal load/store cache hints on the
// dominant streaming pass so the 1MB mask plane stays L2-resident.
// ---------------------------------------------------------------------------

typedef float v2f __attribute__((ext_vector_type(2)));
typedef float v4f __attribute__((ext_vector_type(4)));
typedef float v8f __attribute__((ext_vector_type(8)));

#define SH 364
#define SW 280
#define IMG (SH * SW)          // 101920
#define RADIUS 15
#define KW 31
#define OH 512
#define OW 512

// Branchless zero-padded fetch: always load from a clamped (in-bounds)
// address, then select 0 when logically out of range.
__device__ __forceinline__ float padload(const float* __restrict__ p,
                                         int idx, int maxidx, bool ok) {
    const int ci = min(max(idx, 0), maxidx);
    const float v = p[ci];
    return (ok && idx == ci) ? v : 0.0f;
}

__device__ __forceinline__ float uval(const float* __restrict__ u, int j) {
    const int cj = min(max(j, 0), KW - 1);
    const float v = u[cj];
    return (j == cj) ? v : 0.0f;
}

// ---------------------------------------------------------------------------
// K0: derive separable 1-D kernel u, zero heatmap, scatter keypoints.
// Single block; trivial work.
// ---------------------------------------------------------------------------
__global__ __launch_bounds__(1024) void setup_scatter_kernel(
    const float* __restrict__ landmarks,   // (32, 68, 2)
    const float* __restrict__ weight,      // (1,1,31,31)
    float* __restrict__ u,                 // [31]
    float* __restrict__ hm)                // [364*280]
{
    const int t = threadIdx.x;
    if (t < KW) {
        // k[i][j] = u[i]*u[j]  =>  u[i] = k[i][R] / sqrt(k[R][R])
        u[t] = weight[t * KW + RADIUS] / sqrtf(weight[RADIUS * KW + RADIUS]);
    }
    for (int i = t; i < IMG; i += 1024) hm[i] = 0.0f;
    __syncthreads();

    const float scaling = 364.0f / 512.0f;
    for (int p = t; p < 32 * 96; p += 1024) {
        const int b = p / 96, j = p % 96;
        int idx; float s;
        if      (j <  6) { idx = 30 + j;        s = 1.0f; }  // p0
        else if (j < 12) { idx = 30 + (j - 6);  s = 1.5f; }  // p01
        else if (j < 25) { idx =  2 + (j - 12); s = 0.7f; }  // p12
        else if (j < 38) { idx =  2 + (j - 25); s = 0.9f; }  // p13
        else if (j < 51) { idx =  2 + (j - 38); s = 0.5f; }  // p11
        else if (j < 71) { idx = 48 + (j - 51); s = 1.0f; }  // p2
        else if (j < 76) { idx =  6 + (j - 71); s = 1.0f; }  // p3
        else             { idx = 48 + (j - 76); s = 1.3f; }  // p21
        const float* lmb = landmarks + b * 68 * 2;
        const float cx = lmb[30 * 2 + 0] * scaling;
        const float cy = lmb[30 * 2 + 1] * scaling;
        const float px = lmb[idx * 2 + 0] * scaling;
        const float py = lmb[idx * 2 + 1] * scaling;
        const float kx = (px - cx) * s + cx;
        const float ky = (py - cy) * s + cy;
        const int ix = (int)rintf(fminf(fmaxf(kx, 0.0f), (float)(SW - 1)));
        const int iy = (int)rintf(fminf(fmaxf(ky, 0.0f), (float)(SH - 1)));
        hm[iy * SW + ix] = 1.0f;  // all racers write 1.0
    }
}

// ---------------------------------------------------------------------------
// Horizontal 1-D conv (SAME, zero-pad) on a 16x16 tile per wave.
// D(16x16) = A(16x48) x B(48x16):
//   A[m,k] = src[rbase+m, cbase-15+k]   (image patch)
//   B[k,n] = u[k-n] if 0<=k-n<=30 else 0 (banded Toeplitz)
// f32 WMMA A layout: lane m=L%16; VGPR0/1 = K = 2*half / 2*half+1.
// ---------------------------------------------------------------------------
__global__ __launch_bounds__(32) void rowconv_wmma_kernel(
    const float* __restrict__ src, float* __restrict__ dst,
    const float* __restrict__ u)
{
    const int rbase = blockIdx.y * 16;
    const int cbase = blockIdx.x * 16;
    const int lane  = threadIdx.x & 31;
    const int half  = lane >> 4;
    const int mn    = lane & 15;           // m for A / n for B,D

    const int  r    = min(rbase + mn, SH - 1);  // clamped row (branchless)
    const bool rok  = (rbase + mn) < SH;
    const float* srow = src + r * SW;

    v8f acc = {};
#pragma unroll
    for (int kk = 0; kk < 12; ++kk) {
        const int k0 = kk * 4 + half * 2;
        // A fragment (patch), branchless padded loads
        v2f a;
        const int c0 = cbase - RADIUS + k0;
        a.x = padload(srow, c0,     SW - 1, rok);
        a.y = padload(srow, c0 + 1, SW - 1, rok);
        // B fragment (Toeplitz of u)
        v2f b;
        b.x = uval(u, k0 - mn);
        b.y = uval(u, k0 + 1 - mn);
        acc = __builtin_amdgcn_wmma_f32_16x16x4_f32(
            false, a, false, b, (short)0, acc, false, false);
    }
#pragma unroll
    for (int v = 0; v < 8; ++v) {
        const int rr = rbase + v + 8 * half;
        const int cc = cbase + mn;
        if (rr < SH && cc < SW) dst[rr * SW + cc] = acc[v];
    }
}

// ---------------------------------------------------------------------------
// Vertical 1-D conv, optionally followed by w2 = (w1 > 1e-7) ? 1 : w1.
// D(16x16) = A(16x48) x B(48x16):
//   A[m,k] = u[k-m] (Toeplitz)       B[k,n] = src[rbase-15+k, cbase+n]
// ---------------------------------------------------------------------------
template <bool THRESH>
__global__ __launch_bounds__(32) void colconv_wmma_kernel(
    const float* __restrict__ src, float* __restrict__ dst,
    const float* __restrict__ u)
{
    const int rbase = blockIdx.y * 16;
    const int cbase = blockIdx.x * 16;
    const int lane  = threadIdx.x & 31;
    const int half  = lane >> 4;
    const int mn    = lane & 15;

    const int  cc  = min(cbase + mn, SW - 1);   // clamped col (branchless)
    const bool cok = (cbase + mn) < SW;
    const float* scol = src + cc;

    v8f acc = {};
#pragma unroll
    for (int kk = 0; kk < 12; ++kk) {
        const int k0 = kk * 4 + half * 2;
        // A fragment (Toeplitz of u)
        v2f a;
        a.x = uval(u, k0 - mn);
        a.y = uval(u, k0 + 1 - mn);
        // B fragment (patch), branchless padded loads along rows
        v2f b;
        const int r0 = rbase - RADIUS + k0;
        const int r0c = min(max(r0, 0), SH - 1);
        const int r1c = min(max(r0 + 1, 0), SH - 1);
        const float bv0 = scol[r0c * SW];
        const float bv1 = scol[r1c * SW];
        b.x = (cok && r0     == r0c) ? bv0 : 0.0f;
        b.y = (cok && r0 + 1 == r1c) ? bv1 : 0.0f;
        acc = __builtin_amdgcn_wmma_f32_16x16x4_f32(
            false, a, false, b, (short)0, acc, false, false);
    }
#pragma unroll
    for (int v = 0; v < 8; ++v) {
        const int rr = rbase + v + 8 * half;
        const int c2 = cbase + mn;
        if (rr < SH && c2 < SW) {
            float val = acc[v];
            if (THRESH) val = (val > 1e-7f) ? 1.0f : val;
            dst[rr * SW + c2] = val;
        }
    }
}

// ---------------------------------------------------------------------------
// K5: bilinear resize (half-pixel centers, edge clamp) 364x280 -> 512x512,
// storing the final multiplicative factor (1 - mask).
// ---------------------------------------------------------------------------
__global__ __launch_bounds__(256) void resize_factor_kernel(
    const float* __restrict__ w3, float* __restrict__ f)
{
    const int idx = blockIdx.x * blockDim.x + threadIdx.x;
    if (idx >= OH * OW) return;
    const int y = idx >> 9;
    const int x = idx & (OW - 1);
    const float fy = (y + 0.5f) * ((float)SH / (float)OH) - 0.5f;
    const float fx = (x + 0.5f) * ((float)SW / (float)OW) - 0.5f;
    const int y0 = (int)floorf(fy);
    const int x0 = (int)floorf(fx);
    const float wy = fy - (float)y0;
    const float wx = fx - (float)x0;
    const int y0c = min(max(y0, 0), SH - 1);
    const int y1c = min(max(y0 + 1, 0), SH - 1);
    const int x0c = min(max(x0, 0), SW - 1);
    const int x1c = min(max(x0 + 1, 0), SW - 1);
    const float v00 = w3[y0c * SW + x0c];
    const float v01 = w3[y0c * SW + x1c];
    const float v10 = w3[y1c * SW + x0c];
    const float v11 = w3[y1c * SW + x1c];
    const float m = (1.0f - wy) * ((1.0f - wx) * v00 + wx * v01)
                  +         wy  * ((1.0f - wx) * v10 + wx * v11);
    f[idx] = 1.0f - m;
}

// ---------------------------------------------------------------------------
// K6: streaming out = f[y,x] * in. float4-vectorized. Non-temporal hints on
// the 100MB+100MB in/out streams keep them from evicting the 1MB factor
// plane from L2; factor loads use the default (regular temporal) policy.
// ---------------------------------------------------------------------------
__global__ __launch_bounds__(256) void apply_kernel(
    const v4f* __restrict__ in, const v4f* __restrict__ f,
    v4f* __restrict__ out, int n4)
{
    const int stride = gridDim.x * blockDim.x;
    for (int i = blockIdx.x * blockDim.x + threadIdx.x; i < n4; i += stride) {
        const v4f a = __builtin_nontemporal_load(&in[i]);
        const v4f m = f[i & 65535];   // 512*512/4 float4s per plane (L2-hot)
        __builtin_nontemporal_store(a * m, &out[i]);
    }
}

extern "C" void kernel_launch(void* const* d_in, const int* in_sizes, int n_in,
                              void* d_out, int out_size, void* d_ws, size_t ws_size,
                              hipStream_t stream) {
    (void)in_sizes; (void)n_in; (void)out_size; (void)ws_size;
    const float* input_face = (const float*)d_in[0];   // 32*3*512*512
    const float* landmarks  = (const float*)d_in[1];   // 32*68*2
    const float* weight     = (const float*)d_in[2];   // 31*31

    // Workspace layout (floats): u[32] | bufA | bufB | bufC | factor(512*512)
    float* u    = (float*)d_ws;
    float* bufA = u + 32;
    float* bufB = bufA + IMG;
    float* bufC = bufB + IMG;
    float* fact = bufC + IMG;

    // K0: u, zero heatmap, scatter points -> bufA (= hm)
    setup_scatter_kernel<<<1, 1024, 0, stream>>>(landmarks, weight, u, bufA);

    const dim3 btile((SW + 15) / 16, (SH + 15) / 16);  // 18 x 23 tiles
    // blur1: hm -> t1 -> w2 (threshold)
    rowconv_wmma_kernel<<<btile, 32, 0, stream>>>(bufA, bufB, u);
    colconv_wmma_kernel<true><<<btile, 32, 0, stream>>>(bufB, bufC, u);
    // blur2: w2 -> t2 -> w3
    rowconv_wmma_kernel<<<btile, 32, 0, stream>>>(bufC, bufA, u);
    colconv_wmma_kernel<false><<<btile, 32, 0, stream>>>(bufA, bufB, u);

    // resize + (1 - mask)
    resize_factor_kernel<<<(OH * OW + 255) / 256, 256, 0, stream>>>(bufB, fact);

    // streaming apply
    const int n4 = 32 * 3 * OH * OW / 4;
    apply_kernel<<<4096, 256, 0, stream>>>(
        (const v4f*)input_face, (const v4f*)fact, (v4f*)d_out, n4);
}